// GCNNet_25821343384095
// MI455X (gfx1250) — compile-verified
//
#include <hip/hip_runtime.h>
#include <hip/hip_bf16.h>

typedef __attribute__((ext_vector_type(2))) float v2f;
typedef __attribute__((ext_vector_type(8))) float v8f;

#define EMB 128           // K dimension for both layers (EMB == HID == 128)
#define XPITCH 132        // padded LDS pitch for X tile (breaks 64-bank conflicts)
#define KSTEPS (EMB / 4)  // 32 k-steps of 4 per WMMA

#if defined(__has_builtin)
#if __has_builtin(__builtin_amdgcn_sched_barrier)
#define SCHED_FENCE() __builtin_amdgcn_sched_barrier(0)
#endif
#endif
#ifndef SCHED_FENCE
#define SCHED_FENCE()
#endif

// ---------------------------------------------------------------- utilities
__global__ void fill_kernel(float* __restrict__ p, float v, long long n) {
    long long i = (long long)blockIdx.x * blockDim.x + threadIdx.x;
    long long stride = (long long)gridDim.x * blockDim.x;
    for (; i < n; i += stride) p[i] = v;
}

__global__ void deg_accum_kernel(const int* __restrict__ dst, float* __restrict__ deg, int e) {
    int i = blockIdx.x * blockDim.x + threadIdx.x;
    if (i < e) atomicAdd(&deg[dst[i]], 1.0f);
}

__global__ void rsqrt_ip_kernel(float* __restrict__ d, int n) {
    int i = blockIdx.x * blockDim.x + threadIdx.x;
    if (i < n) d[i] = rsqrtf(d[i]);
}

// ---------------------------------------------------------------- WMMA GEMM
// H[n, HOUT] = X[n, 128] @ W[128, HOUT], fp32 exact via V_WMMA_F32_16X16X4_F32.
// Block: 256 threads (8 wave32), 128 rows per block. W + padded X tile in LDS.
// Per n-tile: burst-load all 32 B fragments into registers (pinned by a
// sched_barrier so the scheduler cannot re-serialize them against the MMAs),
// then run the 32-deep WMMA accumulation chain behind progressive DS waits.
template <int HOUT>
__global__ __launch_bounds__(256) void gemm_wmma_kernel(const float* __restrict__ X,
                                                        const float* __restrict__ W,
                                                        float* __restrict__ H, int nrows) {
    extern __shared__ float smem[];
    float* sW = smem;                 // [128][HOUT], same layout as global
    float* sX = smem + EMB * HOUT;    // [128][XPITCH]

    const int tid = threadIdx.x;
    const int rowBase = blockIdx.x * 128;

    // stage W (contiguous copy)
    for (int i = tid * 4; i < EMB * HOUT; i += 256 * 4) {
        *(float4*)&sW[i] = *(const float4*)&W[i];
    }
    // stage X tile, zero-padded past nrows
    for (int i = tid * 4; i < 128 * EMB; i += 256 * 4) {
        int r = i >> 7, c = i & 127;
        int gr = rowBase + r;
        float4 v = make_float4(0.f, 0.f, 0.f, 0.f);
        if (gr < nrows) v = *(const float4*)&X[(long long)gr * EMB + c];
        *(float4*)&sX[r * XPITCH + c] = v;
    }
    __syncthreads();

    const int wave = tid >> 5;
    const int lane = tid & 31;
    const int nl = lane & 15;      // N (and M) index within 16
    const int kh = lane >> 4;      // lane half selects K pair (A) / output row half (D)
    const int wr = wave * 16;      // this wave's 16-row slice of the tile

    // preload all A fragments for this wave's 16 rows (whole K range)
    // A 16x4 f32: lanes0-15 -> K=k,k+1 ; lanes16-31 -> K=k+2,k+3 (M = lane%16)
    const float* xrow = &sX[(wr + nl) * XPITCH + 2 * kh];
    v2f afrag[KSTEPS];
#pragma unroll
    for (int ks = 0; ks < KSTEPS; ++ks) afrag[ks] = *(const v2f*)&xrow[4 * ks];
    SCHED_FENCE();

    for (int nt = 0; nt < HOUT / 16; ++nt) {
        const float* wcol = &sW[nt * 16 + nl];
        // burst-load B fragments (B 4x16 f32, mirrored lane layout, N = lane%16)
        v2f bfrag[KSTEPS];
#pragma unroll
        for (int ks = 0; ks < KSTEPS; ++ks) {
            const int kk = 4 * ks + 2 * kh;
            v2f b;
            b.x = wcol[kk * HOUT];
            b.y = wcol[(kk + 1) * HOUT];
            bfrag[ks] = b;
        }
        SCHED_FENCE();  // keep the DS burst ahead of the MMA chain
        // back-to-back accumulation chain (waits become progressive dscnt<=n)
        v8f acc = {};
#pragma unroll
        for (int ks = 0; ks < KSTEPS; ++ks) {
            acc = __builtin_amdgcn_wmma_f32_16x16x4_f32(
                false, afrag[ks], false, bfrag[ks], (short)0, acc, false, false);
        }
        SCHED_FENCE();
        // D 16x16 f32: VGPR j -> M = (lane<16 ? j : 8+j), N = lane%16
#pragma unroll
        for (int j = 0; j < 8; ++j) {
            int r = rowBase + wr + kh * 8 + j;
            if (r < nrows) H[(long long)r * HOUT + nt * 16 + nl] = acc[j];
        }
    }
}

// ---------------------------------------------------------------- edge scatter
// agg[dst] += dinv[src]*dinv[dst] * h[src].  One wave per edge; lane covers
// HOUT/32 contiguous features. Atomic f32 adds resolve in L2 (all buffers L2-resident).
template <int HOUT>
__global__ __launch_bounds__(256) void scatter_kernel(const int* __restrict__ src,
                                                      const int* __restrict__ dst,
                                                      const float* __restrict__ dinv,
                                                      const float* __restrict__ h,
                                                      float* __restrict__ agg, int e) {
    int gid = blockIdx.x * blockDim.x + threadIdx.x;
    int edge = gid >> 5;
    int lane = gid & 31;
    if (edge >= e) return;
    int s = src[edge], d = dst[edge];
    float w = dinv[s] * dinv[d];
    constexpr int V = HOUT / 32;
    const float* hs = &h[(long long)s * HOUT + lane * V];
    float* ad = &agg[(long long)d * HOUT + lane * V];
    if constexpr (V == 4) {
        float4 v = *(const float4*)hs;                 // global_load_b128
        atomicAdd(&ad[0], w * v.x);
        atomicAdd(&ad[1], w * v.y);
        atomicAdd(&ad[2], w * v.z);
        atomicAdd(&ad[3], w * v.w);
    } else {
        float2 v = *(const float2*)hs;                 // global_load_b64
        atomicAdd(&ad[0], w * v.x);
        atomicAdd(&ad[1], w * v.y);
    }
}

// out = relu(agg + dinv^2*h + b)   (self-loop folded into epilogue; in-place over agg)
__global__ void bias_relu_kernel(float* __restrict__ agg, const float* __restrict__ h,
                                 const float* __restrict__ dinv, const float* __restrict__ b,
                                 long long total, int hdim) {
    long long i = (long long)blockIdx.x * blockDim.x + threadIdx.x;
    if (i >= total) return;
    int row = (int)(i / hdim), col = (int)(i % hdim);
    float di = dinv[row];
    float v = agg[i] + di * di * h[i] + b[col];
    agg[i] = fmaxf(v, 0.0f);
}

// out += dinv^2*h + b   (final layer epilogue, in-place over scattered d_out)
__global__ void final_add_kernel(float* __restrict__ out, const float* __restrict__ h,
                                 const float* __restrict__ dinv, const float* __restrict__ b,
                                 long long total, int hdim) {
    long long i = (long long)blockIdx.x * blockDim.x + threadIdx.x;
    if (i >= total) return;
    int row = (int)(i / hdim), col = (int)(i % hdim);
    float di = dinv[row];
    out[i] = out[i] + di * di * h[i] + b[col];
}

// ---------------------------------------------------------------- launcher
extern "C" void kernel_launch(void* const* d_in, const int* in_sizes, int n_in,
                              void* d_out, int out_size, void* d_ws, size_t ws_size,
                              hipStream_t stream) {
    const int*   edge = (const int*)d_in[0];
    const float* x0   = (const float*)d_in[1];
    const float* W1   = (const float*)d_in[2];
    const float* b1   = (const float*)d_in[3];
    const float* W2   = (const float*)d_in[4];
    const float* b2   = (const float*)d_in[5];
    float* out = (float*)d_out;

    const int E    = in_sizes[0] / 2;
    const int N    = in_sizes[1] / EMB;   // 100000
    const int HID  = in_sizes[3];         // 128
    const int REPR = in_sizes[5];         // 64
    const int* src = edge;
    const int* dst = edge + E;

    // workspace layout (all fp32): dinv[N] | h[N*HID] (reused for h2) | agg[N*HID] (reused as x1)
    float* dinv = (float*)d_ws;
    float* h    = dinv + N;
    float* agg  = h + (size_t)N * HID;

    const int T = 256;
    long long nh = (long long)N * HID;
    long long nr = (long long)N * REPR;

    // degrees: start at 1 (self-loop), accumulate edge dst, then rsqrt in place
    fill_kernel<<<512, T, 0, stream>>>(dinv, 1.0f, N);
    deg_accum_kernel<<<(E + T - 1) / T, T, 0, stream>>>(dst, dinv, E);
    rsqrt_ip_kernel<<<(N + T - 1) / T, T, 0, stream>>>(dinv, N);

    // ---- layer 1: h = x0 @ W1 ; agg = scatter(norm*h) ; x1 = relu(agg + dinv^2*h + b1)
    size_t smem1 = (size_t)(EMB * HID + 128 * XPITCH) * sizeof(float);
    gemm_wmma_kernel<128><<<(N + 127) / 128, T, smem1, stream>>>(x0, W1, h, N);
    fill_kernel<<<2048, T, 0, stream>>>(agg, 0.0f, nh);
    scatter_kernel<128><<<(int)(((long long)E * 32 + T - 1) / T), T, 0, stream>>>(
        src, dst, dinv, h, agg, E);
    bias_relu_kernel<<<(int)((nh + T - 1) / T), T, 0, stream>>>(agg, h, dinv, b1, nh, HID);

    // ---- layer 2: h2 = x1 @ W2 (into h region) ; d_out = scatter + self-loop + b2
    size_t smem2 = (size_t)(EMB * REPR + 128 * XPITCH) * sizeof(float);
    gemm_wmma_kernel<64><<<(N + 127) / 128, T, smem2, stream>>>(agg, W2, h, N);
    fill_kernel<<<2048, T, 0, stream>>>(out, 0.0f, nr);
    scatter_kernel<64><<<(int)(((long long)E * 32 + T - 1) / T), T, 0, stream>>>(
        src, dst, dinv, h, out, E);
    final_add_kernel<<<(int)((nr + T - 1) / T), T, 0, stream>>>(out, h, dinv, b2, nr, REPR);
}